// DiagonalMatrixModel_62801011802137
// MI455X (gfx1250) — compile-verified
//
#include <hip/hip_runtime.h>

// out[b, j] = x[b, j] * diagonal[j]   (x @ diag(diagonal))
//
// Pure streaming kernel: 268 MB moved once, 33.5 MFLOP. Memory-bound floor on
// MI455X HBM (23.3 TB/s) ~= 11.5 us. Design goals:
//   - b128 (float4) loads/stores, coalesced across wave32 lanes
//   - non-temporal (TH=NT) on the 256 MiB x/out streams (no L2 pollution;
//     diagonal stays regular-temporal and is L2-resident at 16 KiB)
//   - per-thread diagonal values loaded once, reused across 16 rows
//   - 4 rows in flight per iteration + gfx1250 global_prefetch for next tile

typedef __attribute__((ext_vector_type(4))) float v4f;

#ifndef DIAG_SIZE
#define DIAG_SIZE 4096
#endif
#ifndef DIAG_BATCH
#define DIAG_BATCH 8192
#endif

__global__ __launch_bounds__(256) void diag_scale_stream_kernel(
    const float* __restrict__ x,
    const float* __restrict__ diag,
    float* __restrict__ out)
{
    constexpr int SIZE = DIAG_SIZE;           // columns
    constexpr int BATCH = DIAG_BATCH;         // rows
    constexpr int VEC = 4;                    // float4 per lane
    constexpr int THREADS = 256;              // 8 waves (wave32)
    constexpr int COLS_PER_BLOCK = THREADS * VEC;     // 1024 floats
    constexpr int COL_SEGS = SIZE / COLS_PER_BLOCK;   // 4 column segments
    constexpr int ROWS_PER_ITER = 4;

    const int seg       = blockIdx.x % COL_SEGS;
    const int rowBlock  = blockIdx.x / COL_SEGS;
    const int nRowBlks  = gridDim.x / COL_SEGS;

    const int col = seg * COLS_PER_BLOCK + (int)threadIdx.x * VEC;

    // Load this lane's 4 diagonal values once; reused for every row this
    // block touches. Regular-temporal: shared by all 512 row-blocks of this
    // column segment, so let it live in WGP$/L2.
    const v4f d = *(const v4f*)(diag + col);

    const int rowStride = nRowBlks * ROWS_PER_ITER;

    for (int row = rowBlock * ROWS_PER_ITER; row < BATCH; row += rowStride) {
        const float* __restrict__ xp = x   + (size_t)row * SIZE + col;
        float*       __restrict__ op = out + (size_t)row * SIZE + col;

        // Prefetch the next row-tile this block will process (speculative;
        // dropped silently if past the end). Lowers to global_prefetch_b8.
        {
            const size_t nextOff = (size_t)rowStride * SIZE;
            __builtin_prefetch(xp + nextOff, 0, 0);
        }

        // 4 outstanding b128 NT loads
        v4f a0 = __builtin_nontemporal_load((const v4f*)(xp + 0 * SIZE));
        v4f a1 = __builtin_nontemporal_load((const v4f*)(xp + 1 * SIZE));
        v4f a2 = __builtin_nontemporal_load((const v4f*)(xp + 2 * SIZE));
        v4f a3 = __builtin_nontemporal_load((const v4f*)(xp + 3 * SIZE));

        a0 *= d;
        a1 *= d;
        a2 *= d;
        a3 *= d;

        // 4 outstanding b128 NT stores (write-once stream)
        __builtin_nontemporal_store(a0, (v4f*)(op + 0 * SIZE));
        __builtin_nontemporal_store(a1, (v4f*)(op + 1 * SIZE));
        __builtin_nontemporal_store(a2, (v4f*)(op + 2 * SIZE));
        __builtin_nontemporal_store(a3, (v4f*)(op + 3 * SIZE));
    }
}

extern "C" void kernel_launch(void* const* d_in, const int* in_sizes, int n_in,
                              void* d_out, int out_size, void* d_ws, size_t ws_size,
                              hipStream_t stream) {
    (void)in_sizes; (void)n_in; (void)out_size; (void)d_ws; (void)ws_size;

    const float* x    = (const float*)d_in[0];   // [BATCH, SIZE] f32
    const float* diag = (const float*)d_in[1];   // [SIZE] f32
    float*       out  = (float*)d_out;           // [BATCH, SIZE] f32

    constexpr int THREADS = 256;
    constexpr int COL_SEGS = DIAG_SIZE / (THREADS * 4);  // 4
    constexpr int ROW_BLOCKS = 512;                      // 4*512 = 2048 blocks
    // Each block: 4 rows/iter * (8192 / (512*4)) = 4 iterations over 4 KiB-wide tile.

    dim3 grid(COL_SEGS * ROW_BLOCKS);
    dim3 block(THREADS);
    diag_scale_stream_kernel<<<grid, block, 0, stream>>>(x, diag, out);
}